// LightGCN_15049565405254
// MI455X (gfx1250) — compile-verified
//
#include <hip/hip_runtime.h>
#include <hip/hip_bf16.h>

#define NNODES 100000
#define NEDGES 1200000
#define HID 64
#define NLAYERS 3
#define GAMMA_C (-0.45f)
#define ZETA_C  (1.05f)

typedef float v2f __attribute__((ext_vector_type(2)));
typedef float v8f __attribute__((ext_vector_type(8)));

// ---------------------------------------------------------------------------
// p = relu(x @ W1 + b1) @ W2 (+ bias2 if has_bias2)
// One 16-node tile per wave, 4 waves per 128-thread block.
// Uses V_WMMA_F32_16X16X4_F32 (exact fp32).
// ---------------------------------------------------------------------------
__global__ __launch_bounds__(128) void node_transform_kernel(
    const float* __restrict__ x, const float* __restrict__ W1,
    const float* __restrict__ b1, const float* __restrict__ W2,
    const float* __restrict__ bias2, int has_bias2,
    float* __restrict__ p, int n_nodes)
{
    __shared__ float w1s[64 * 64];
    __shared__ float w2s[64 * 64];
    __shared__ float gs[4 * 16 * 68];   // per-wave staging, stride 68 (no bank conflicts)

    for (int i = threadIdx.x; i < 64 * 64; i += 128) {
        w1s[i] = W1[i];
        w2s[i] = W2[i];
    }
    __syncthreads();

    const int wave = threadIdx.x >> 5;
    const int lane = threadIdx.x & 31;
    const int mn    = lane & 15;          // A-row / B-col / C-col index
    const int kh    = (lane >> 4) << 1;   // 0 or 2: K sub-offset within 4-chunk
    const int mbase = (lane >> 4) << 3;   // C/D layout: lanes16-31 hold M+8
    float* gt = gs + wave * (16 * 68);

    const int tile = blockIdx.x * 4 + wave;
    const int row0 = tile * 16;
    if (row0 >= n_nodes) return;

    int ri = row0 + mn;
    if (ri >= n_nodes) ri = n_nodes - 1;  // dup-read guard (store is guarded)
    const float* xr = x + (size_t)ri * HID;

    const v8f vz = {0.f, 0.f, 0.f, 0.f, 0.f, 0.f, 0.f, 0.f};
    v8f acc[4] = {vz, vz, vz, vz};

    // ---- GEMM1: x_tile(16x64) @ W1(64x64) ----
    #pragma unroll
    for (int kk = 0; kk < 16; ++kk) {
        const int k = kk * 4 + kh;
        v2f a; a.x = xr[k]; a.y = xr[k + 1];
        #pragma unroll
        for (int nt = 0; nt < 4; ++nt) {
            v2f b;
            b.x = w1s[k * 64 + nt * 16 + mn];
            b.y = w1s[(k + 1) * 64 + nt * 16 + mn];
            acc[nt] = __builtin_amdgcn_wmma_f32_16x16x4_f32(
                false, a, false, b, (short)0, acc[nt], false, false);
        }
    }

    // bias + relu, stage to LDS in row-major [M][K] (stride 68)
    #pragma unroll
    for (int nt = 0; nt < 4; ++nt) {
        const float bv = b1[nt * 16 + mn];
        #pragma unroll
        for (int i = 0; i < 8; ++i) {
            float v = acc[nt][i] + bv;
            v = v > 0.f ? v : 0.f;
            gt[(mbase + i) * 68 + nt * 16 + mn] = v;
        }
    }

    // ---- GEMM2: g_tile(16x64) @ W2(64x64) ----
    v8f acc2[4] = {vz, vz, vz, vz};
    #pragma unroll
    for (int kk = 0; kk < 16; ++kk) {
        const int k = kk * 4 + kh;
        v2f a; a.x = gt[mn * 68 + k]; a.y = gt[mn * 68 + k + 1];
        #pragma unroll
        for (int nt = 0; nt < 4; ++nt) {
            v2f b;
            b.x = w2s[k * 64 + nt * 16 + mn];
            b.y = w2s[(k + 1) * 64 + nt * 16 + mn];
            acc2[nt] = __builtin_amdgcn_wmma_f32_16x16x4_f32(
                false, a, false, b, (short)0, acc2[nt], false, false);
        }
    }

    // store D (C-layout) + optional bias2
    #pragma unroll
    for (int nt = 0; nt < 4; ++nt) {
        const float b2v = has_bias2 ? bias2[nt * 16 + mn] : 0.f;
        #pragma unroll
        for (int i = 0; i < 8; ++i) {
            const int m = mbase + i;
            const int r = row0 + m;
            if (r < n_nodes)
                p[(size_t)r * HID + nt * 16 + mn] = acc2[nt][i] + b2v;
        }
    }
}

// ---------------------------------------------------------------------------
// per-edge attention scalar: one edge per 16-lane group (float4 per lane)
// mask[e] = clip(sigmoid(relu(p1[row]+p2[col]) . w2 + b2) * (Z-G) + G, 0, 1)
// rowsum[row] += mask[e]
// ---------------------------------------------------------------------------
__global__ __launch_bounds__(256) void edge_att_kernel(
    const float* __restrict__ p1, const float* __restrict__ p2,
    const float* __restrict__ w2, const float* __restrict__ b2,
    const int* __restrict__ row, const int* __restrict__ col,
    float* __restrict__ mask, float* __restrict__ rowsum, int n_edges)
{
    const int e = blockIdx.x * 16 + (threadIdx.x >> 4);
    if (e >= n_edges) return;
    const int l4 = (threadIdx.x & 15) * 4;

    const int r = row[e];
    const int c = col[e];
    const float4 a = *(const float4*)(p1 + (size_t)r * HID + l4);
    const float4 b = *(const float4*)(p2 + (size_t)c * HID + l4);
    const float4 w = *(const float4*)(w2 + l4);

    const float t0 = fmaxf(a.x + b.x, 0.f);
    const float t1 = fmaxf(a.y + b.y, 0.f);
    const float t2 = fmaxf(a.z + b.z, 0.f);
    const float t3 = fmaxf(a.w + b.w, 0.f);
    float s = t0 * w.x + t1 * w.y + t2 * w.z + t3 * w.w;

    s += __shfl_xor(s, 8, 32);
    s += __shfl_xor(s, 4, 32);
    s += __shfl_xor(s, 2, 32);
    s += __shfl_xor(s, 1, 32);

    if ((threadIdx.x & 15) == 0) {
        const float la = s + b2[0];
        const float sg = 1.f / (1.f + __expf(-la));
        float m = sg * (ZETA_C - GAMMA_C) + GAMMA_C;
        m = fminf(fmaxf(m, 0.f), 1.f);
        mask[e] = m;
        atomicAdd(rowsum + r, m);
    }
}

// d_inv_sqrt = clip((rowsum + 1e-6)^-0.5, 0, 10)
__global__ __launch_bounds__(256) void dinv_kernel(
    const float* __restrict__ rowsum, float* __restrict__ dis, int n)
{
    const int i = blockIdx.x * 256 + threadIdx.x;
    if (i < n) {
        const float s = rowsum[i] + 1e-6f;
        dis[i] = fminf(rsqrtf(s), 10.f);
    }
}

// ---------------------------------------------------------------------------
// SpMM scatter: xn[row] += mask*dis[row]*dis[col] * x[col]
// one edge per 16-lane group, 4 f32 atomics per lane (L2-resident)
// ---------------------------------------------------------------------------
__global__ __launch_bounds__(256) void edge_spmm_kernel(
    const float* __restrict__ x, const float* __restrict__ mask,
    const float* __restrict__ dis,
    const int* __restrict__ row, const int* __restrict__ col,
    float* __restrict__ xn, int n_edges)
{
    const int e = blockIdx.x * 16 + (threadIdx.x >> 4);
    if (e >= n_edges) return;
    const int l4 = (threadIdx.x & 15) * 4;

    const int r = row[e];
    const int c = col[e];
    const float v = mask[e] * dis[r] * dis[c];
    const float4 xc = *(const float4*)(x + (size_t)c * HID + l4);
    float* dst = xn + (size_t)r * HID + l4;
    atomicAdd(dst + 0, v * xc.x);
    atomicAdd(dst + 1, v * xc.y);
    atomicAdd(dst + 2, v * xc.z);
    atomicAdd(dst + 3, v * xc.w);
}

__global__ __launch_bounds__(256) void copy2_kernel(
    const float* __restrict__ src, float* __restrict__ d0,
    float* __restrict__ d1, int n)
{
    const int i = blockIdx.x * 256 + threadIdx.x;
    if (i < n) { const float v = src[i]; d0[i] = v; d1[i] = v; }
}

__global__ __launch_bounds__(256) void fill0_kernel(float* __restrict__ p, int n)
{
    const int i = blockIdx.x * 256 + threadIdx.x;
    if (i < n) p[i] = 0.f;
}

__global__ __launch_bounds__(256) void accum_kernel(
    float* __restrict__ out, const float* __restrict__ xn, int n)
{
    const int i = blockIdx.x * 256 + threadIdx.x;
    if (i < n) out[i] += xn[i];
}

extern "C" void kernel_launch(void* const* d_in, const int* in_sizes, int n_in,
                              void* d_out, int out_size, void* d_ws, size_t ws_size,
                              hipStream_t stream) {
    (void)in_sizes; (void)n_in; (void)out_size; (void)ws_size;

    const float* feat = (const float*)d_in[0];
    const float* nbW  = (const float*)d_in[1];
    const float* nbb  = (const float*)d_in[2];
    const float* sW   = (const float*)d_in[3];
    const float* sb   = (const float*)d_in[4];
    const float* a1W  = (const float*)d_in[5];   // [L, 128, 64]
    const float* a1b  = (const float*)d_in[6];
    const float* a2W  = (const float*)d_in[7];   // [L, 64, 1]
    const float* a2b  = (const float*)d_in[8];   // [L, 1]
    const int*   row  = (const int*)d_in[9];
    const int*   col  = (const int*)d_in[10];
    float* out = (float*)d_out;

    const int NH = NNODES * HID;   // 6,400,000
    float* ws = (float*)d_ws;
    float* xa     = ws; ws += NH;
    float* xb     = ws; ws += NH;
    float* p1     = ws; ws += NH;
    float* p2     = ws; ws += NH;
    float* maskb  = ws; ws += NEDGES;
    float* rowsum = ws; ws += NNODES;
    float* dis    = ws; ws += NNODES;

    const int gNH    = (NH + 255) / 256;
    const int gNode  = (NNODES + 255) / 256;
    const int gTile  = ((NNODES + 15) / 16 + 3) / 4;   // 4 tiles per 128-thread block
    const int gEdge  = (NEDGES + 15) / 16;

    // out = features; x = features
    copy2_kernel<<<gNH, 256, 0, stream>>>(feat, xa, out, NH);

    float* xcur = xa;
    float* xnew = xb;
    for (int l = 0; l < NLAYERS; ++l) {
        fill0_kernel<<<gNode, 256, 0, stream>>>(rowsum, NNODES);
        fill0_kernel<<<gNH, 256, 0, stream>>>(xnew, NH);

        // p1 = relu(x@nbW + nbb) @ att1W_top + att1_b   (bias baked in)
        node_transform_kernel<<<gTile, 128, 0, stream>>>(
            xcur, nbW + (size_t)l * 4096, nbb + l * 64,
            a1W + (size_t)l * 8192, a1b + l * 64, 1, p1, NNODES);
        // p2 = relu(x@selfW + selfb) @ att1W_bot
        node_transform_kernel<<<gTile, 128, 0, stream>>>(
            xcur, sW + (size_t)l * 4096, sb + l * 64,
            a1W + (size_t)l * 8192 + 4096, a1b, 0, p2, NNODES);

        edge_att_kernel<<<gEdge, 256, 0, stream>>>(
            p1, p2, a2W + l * 64, a2b + l, row, col, maskb, rowsum, NEDGES);

        dinv_kernel<<<gNode, 256, 0, stream>>>(rowsum, dis, NNODES);

        edge_spmm_kernel<<<gEdge, 256, 0, stream>>>(
            xcur, maskb, dis, row, col, xnew, NEDGES);

        accum_kernel<<<gNH, 256, 0, stream>>>(out, xnew, NH);

        float* t = xcur; xcur = xnew; xnew = t;
    }
}